// ConditionalLayer_66099546685946
// MI455X (gfx1250) — compile-verified
//
#include <hip/hip_runtime.h>

typedef float v2f __attribute__((ext_vector_type(2)));
typedef float v4f __attribute__((ext_vector_type(4)));
typedef float v8f __attribute__((ext_vector_type(8)));
typedef int   v4i __attribute__((ext_vector_type(4)));

#define B_ROWS 8192
#define DDIM   1024
#define NCOND  16

#define KC         64                    // k-chunk staged in LDS
#define LDS_STRIDE 68                    // floats per column-row: bank-conflict-free, 16B-aligned
#define BUF_FLOATS (64 * LDS_STRIDE)     // one buffer: 17408 B
#define NCHUNK     (DDIM / KC)

#if defined(__has_builtin)
#if __has_builtin(__builtin_amdgcn_global_load_async_to_lds_b128)
#define HAVE_ASYNC 1
#endif
#endif

typedef __attribute__((address_space(1))) v4i* gv4ip;   // global int4*
typedef __attribute__((address_space(3))) v4i* lv4ip;   // LDS int4*

template <int N>
__device__ __forceinline__ void wait_asynccnt() {
#ifdef HAVE_ASYNC
#if __has_builtin(__builtin_amdgcn_s_wait_asynccnt)
    __builtin_amdgcn_s_wait_asynccnt(N);
#else
    asm volatile("s_wait_asynccnt %0" ::"n"(N));
#endif
#endif
}

// Stage one 64-col x 64-k chunk of W_c^T into LDS (8 x async b128 per thread).
__device__ __forceinline__ void fill_chunk(const float* __restrict__ gbase,
                                           float* lbuf, int tid) {
#pragma unroll
    for (int i = 0; i < 8; ++i) {
        int id  = i * 128 + tid;         // 1024 transfers of 16B = 16KB
        int row = id >> 4;               // output column j = 0..63
        int off = (id & 15) * 4;         // float offset within 64-float row chunk
        const float* g = gbase + (size_t)row * DDIM + off;
        float*       l = lbuf + row * LDS_STRIDE + off;
#ifdef HAVE_ASYNC
        __builtin_amdgcn_global_load_async_to_lds_b128((gv4ip)g, (lv4ip)l, 0, 0);
#else
        *(v4f*)l = *(const v4f*)g;       // sync fallback: load + ds_store
#endif
    }
}

__global__ void init_meta(const float* __restrict__ bias,
                          int* __restrict__ counts,
                          float* __restrict__ bsum) {
    int t = blockIdx.x * blockDim.x + threadIdx.x;
    if (t < DDIM) {
        float s = 0.f;
#pragma unroll
        for (int c = 0; c < NCOND; ++c) s += bias[c * DDIM + t];
        bsum[t] = s;
    }
    if (t < NCOND) counts[t] = 0;
}

__global__ void build_lists(const int* __restrict__ cond,
                            int* __restrict__ lists,
                            int* __restrict__ counts) {
    int r = blockIdx.x * blockDim.x + threadIdx.x;
    if (r < B_ROWS) {
        int c = cond[r];
        int pos = atomicAdd(&counts[c], 1);
        lists[c * B_ROWS + pos] = r;
    }
}

// Block: 4 waves, one condition, one 64-col strip, 64 consecutive row-slots.
// B (W_c^T chunk) staged in LDS once per block, double-buffered via async-to-LDS.
__global__ __launch_bounds__(128)
void grouped_gemm_wmma(const float* __restrict__ x,
                       const float* __restrict__ W,
                       const int*   __restrict__ lists,
                       const int*   __restrict__ counts,
                       const float* __restrict__ bsum,
                       float*       __restrict__ out) {
    __shared__ float smem[2 * BUF_FLOATS];

    const int tid  = threadIdx.x;
    const int lane = tid & 31;
    const int wave = tid >> 5;
    const int c    = blockIdx.y;
    const int cnt  = counts[c];
    const int groupBase = blockIdx.z * 64;          // first row-slot of this block
    if (groupBase >= cnt) return;                   // uniform: whole block exits

    const int tileBase = groupBase + wave * 16;     // this wave's 16-row tile
    const int f0 = blockIdx.x * 64;
    const int lo = lane & 15;
    const int hi = lane >> 4;

    // A fragment gather address (16x4 f32 layout): clamp padded slots to slot 0
    int slotA = tileBase + lo;
    int rowA  = lists[c * B_ROWS + (slotA < cnt ? slotA : 0)];
    const float* aptr = x + (size_t)rowA * DDIM + hi * 2;

    // W_c strip base: rows are output columns f0..f0+63, stride DDIM
    const float* wbase = W + ((size_t)c * DDIM + f0) * DDIM;

    // bias-initialized accumulators (bias is column-constant)
    v8f acc[4];
#pragma unroll
    for (int t = 0; t < 4; ++t) {
        float bs = bsum[f0 + t * 16 + lo];
        acc[t] = (v8f){bs, bs, bs, bs, bs, bs, bs, bs};
    }

    fill_chunk(wbase, smem, tid);                   // prologue: chunk 0 -> buf 0

#pragma unroll 1
    for (int ic = 0; ic < NCHUNK; ++ic) {
        const int kc = ic * KC;
        if (ic + 1 < NCHUNK) {                      // overlap next fill with compute
            fill_chunk(wbase + kc + KC, smem + ((ic + 1) & 1) * BUF_FLOATS, tid);
            wait_asynccnt<8>();                     // chunk ic complete (8 newer in flight)
        } else {
            wait_asynccnt<0>();
        }
        __syncthreads();                            // chunk ic visible to all waves

        const float* lb = smem + (ic & 1) * BUF_FLOATS + lo * LDS_STRIDE + hi * 2;
#pragma unroll
        for (int kk = 0; kk < KC; kk += 4) {
            v2f a = *reinterpret_cast<const v2f*>(aptr + kc + kk);   // global_load_b64
#pragma unroll
            for (int t = 0; t < 4; ++t) {
                v2f bb = *reinterpret_cast<const v2f*>(lb + t * 16 * LDS_STRIDE + kk); // ds_load_b64
                acc[t] = __builtin_amdgcn_wmma_f32_16x16x4_f32(
                    false, a, false, bb, (short)0, acc[t], false, false);
            }
        }
        __syncthreads();                            // done reading buf before its refill
    }

    // scattered, predicated store through the index list
#pragma unroll
    for (int v = 0; v < 8; ++v) {
        int M    = v + hi * 8;
        int slot = tileBase + M;
        if (slot < cnt) {
            int row  = lists[c * B_ROWS + slot];
            float* o = out + (size_t)row * DDIM + f0 + lo;
#pragma unroll
            for (int t = 0; t < 4; ++t) o[t * 16] = acc[t][v];
        }
    }
}

extern "C" void kernel_launch(void* const* d_in, const int* in_sizes, int n_in,
                              void* d_out, int out_size, void* d_ws, size_t ws_size,
                              hipStream_t stream) {
    const float* x    = (const float*)d_in[0];
    const int*   cond = (const int*)d_in[1];
    const float* W    = (const float*)d_in[2];
    const float* bias = (const float*)d_in[3];
    float*       out  = (float*)d_out;

    char*  ws     = (char*)d_ws;
    int*   lists  = (int*)ws;
    int*   counts = (int*)(ws + (size_t)NCOND * B_ROWS * sizeof(int));
    float* bsum   = (float*)(ws + (size_t)NCOND * B_ROWS * sizeof(int) + 64);

    init_meta<<<4, 256, 0, stream>>>(bias, counts, bsum);
    build_lists<<<B_ROWS / 256, 256, 0, stream>>>(cond, lists, counts);
    grouped_gemm_wmma<<<dim3(16, 16, 128), 128, 0, stream>>>(x, W, lists, counts, bsum, out);
}